// SGim_71768903516481
// MI455X (gfx1250) — compile-verified
//
#include <hip/hip_runtime.h>
#include <hip/hip_bf16.h>

// ---------------- problem constants ----------------
constexpr int   kB   = 128;          // graphs
constexpr int   kNPG = 512;          // nodes per graph
constexpr int   kIN  = 32;           // input feature dim
constexpr int   kH   = 64;           // hidden dim
constexpr int   kL   = 3;            // GIN layers
constexpr int   kN   = kB * kNPG;    // 65536 nodes
constexpr int   kE   = kN * 16;      // 1048576 edges
constexpr size_t kNM = (size_t)kN * kH;  // node-matrix elems (4,194,304)

typedef float v2f __attribute__((ext_vector_type(2)));
typedef float v8f __attribute__((ext_vector_type(8)));

// D = A(16x4, f32) * B(4x16, f32) + C(16x16, f32)  -- exact fp32 WMMA
__device__ __forceinline__ v8f wmma4(v2f a, v2f b, v8f c) {
  return __builtin_amdgcn_wmma_f32_16x16x4_f32(false, a, false, b, (short)0, c,
                                               false, false);
}

// =====================================================================
// GEMM: Out[M,64] = (s*A0 [+ A1])[M,K] @ W[64,K]^T + bias
// block = 256 threads (8 waves); each wave -> one 16x16 tile.
// grid.x = M/32  (block covers 32 rows x 64 cols)
// K and HAS_AGG are compile-time: no divergence, W preloaded in VGPRs,
// fully unrolled -> inner body is A-load + v_wmma.
// =====================================================================
template <int K, bool HAS_AGG>
__global__ __launch_bounds__(256) void gemm_rows64_t(
    const float* __restrict__ A0,
    const float* __restrict__ A1,     // used iff HAS_AGG
    const float* __restrict__ epsP,   // used iff HAS_AGG
    int epsIdx,
    const float* __restrict__ W,      // [64,K]
    const float* __restrict__ bias,   // [64]
    float* __restrict__ Out) {
  constexpr int NK = K / 4;
  const int lane  = threadIdx.x & 31;
  const int wave  = threadIdx.x >> 5;          // 0..7
  const int rtile = blockIdx.x * 2 + (wave >> 2);
  const int ctile = wave & 3;
  const int row   = rtile * 16 + (lane & 15);
  const int col   = ctile * 16 + (lane & 15);
  const int khalf = (lane >> 4) * 2;

  // preload this wave's weight columns (B operand) into registers
  const float* wrow = W + (size_t)col * K;
  v2f breg[NK];
#pragma unroll
  for (int i = 0; i < NK; ++i) {
    breg[i].x = wrow[i * 4 + khalf];
    breg[i].y = wrow[i * 4 + khalf + 1];
  }

  v8f acc;
  const float bv = bias[col];
#pragma unroll
  for (int i = 0; i < 8; ++i) acc[i] = bv;

  const float* arow = A0 + (size_t)row * K;
  if constexpr (HAS_AGG) {
    const float s = 1.0f + epsP[epsIdx];
    const float* arow1 = A1 + (size_t)row * K;
#pragma unroll
    for (int i = 0; i < NK; ++i) {
      const int kk = i * 4 + khalf;
      v2f a;
      a.x = s * arow[kk]     + arow1[kk];
      a.y = s * arow[kk + 1] + arow1[kk + 1];
      acc = wmma4(a, breg[i], acc);
    }
  } else {
#pragma unroll
    for (int i = 0; i < NK; ++i) {
      const int kk = i * 4 + khalf;
      v2f a;
      a.x = arow[kk];
      a.y = arow[kk + 1];
      acc = wmma4(a, breg[i], acc);
    }
  }

  const int mbase = rtile * 16 + ((lane >> 4) << 3);
  float* orow = Out + (size_t)mbase * 64 + col;
#pragma unroll
  for (int v = 0; v < 8; ++v) orow[(size_t)v * 64] = acc[v];
}

// =====================================================================
// Gram: NB[b,l] = Xd^T Xd  (Xd = X[b] : [512,64]).  A[m,k]=X[k,m], B[k,n]=X[k,n]
// block = 512 threads (16 waves), one block per graph.
// =====================================================================
__global__ __launch_bounds__(512) void gram_kernel(const float* __restrict__ X,
                                                   float* __restrict__ NB,
                                                   int l) {
  const int b     = blockIdx.x;
  const int lane  = threadIdx.x & 31;
  const int wave  = threadIdx.x >> 5;          // 0..15
  const int rtile = wave >> 2, ctile = wave & 3;
  const int khalf = (lane >> 4) * 2;
  const int rr = rtile * 16 + (lane & 15);
  const int cc = ctile * 16 + (lane & 15);
  const float* Xb = X + (size_t)b * kNPG * kH;

  v8f acc;
#pragma unroll
  for (int i = 0; i < 8; ++i) acc[i] = 0.0f;

#pragma unroll 4
  for (int k0 = 0; k0 < kNPG; k0 += 4) {
    const int kk = k0 + khalf;
    v2f a, bv;
    a.x  = Xb[(size_t)kk * 64 + rr];
    a.y  = Xb[(size_t)(kk + 1) * 64 + rr];
    bv.x = Xb[(size_t)kk * 64 + cc];
    bv.y = Xb[(size_t)(kk + 1) * 64 + cc];
    acc = wmma4(a, bv, acc);
  }

  float* G = NB + ((size_t)b * (kL + 1) + l) * 4096;
  const int mbase = rtile * 16 + ((lane >> 4) << 3);
#pragma unroll
  for (int v = 0; v < 8; ++v) G[(size_t)(mbase + v) * 64 + cc] = acc[v];
}

// =====================================================================
// Gram product: OUT[bl] = NBI[bl] @ NBJ[bl]^T   (64x64 each)
// block = 512 threads (16 waves), grid = B*(L+1)
// =====================================================================
__global__ __launch_bounds__(512) void nbcomb_kernel(
    const float* __restrict__ NBI, const float* __restrict__ NBJ,
    float* __restrict__ OUT) {
  const int bl = blockIdx.x;
  const float* Ai = NBI + (size_t)bl * 4096;
  const float* Bj = NBJ + (size_t)bl * 4096;
  float* O = OUT + (size_t)bl * 4096;

  const int lane  = threadIdx.x & 31;
  const int wave  = threadIdx.x >> 5;
  const int rtile = wave >> 2, ctile = wave & 3;
  const int khalf = (lane >> 4) * 2;
  const int rr = rtile * 16 + (lane & 15);
  const int cc = ctile * 16 + (lane & 15);

  v8f acc;
#pragma unroll
  for (int i = 0; i < 8; ++i) acc[i] = 0.0f;

#pragma unroll
  for (int k0 = 0; k0 < 64; k0 += 4) {
    const int kk = k0 + khalf;
    v2f a, bv;
    a.x  = Ai[(size_t)rr * 64 + kk];       // A[m,j]
    a.y  = Ai[(size_t)rr * 64 + kk + 1];
    bv.x = Bj[(size_t)cc * 64 + kk];       // B[j,n] = NBJ[n,j]
    bv.y = Bj[(size_t)cc * 64 + kk + 1];
    acc = wmma4(a, bv, acc);
  }

  const int mbase = rtile * 16 + ((lane >> 4) << 3);
#pragma unroll
  for (int v = 0; v < 8; ++v) O[(size_t)(mbase + v) * 64 + cc] = acc[v];
}

// =====================================================================
// Edge aggregation: AGG[dst] += X[src]; 16 threads/edge * 4 ch each.
// =====================================================================
__global__ void edge_agg(const float* __restrict__ X,
                         const int* __restrict__ EI,
                         float* __restrict__ AGG) {
  const long long tid = (long long)blockIdx.x * blockDim.x + threadIdx.x;
  const long long e = tid >> 4;
  if (e >= kE) return;
  const int c4 = ((int)tid & 15) << 2;
  const int s = EI[e];
  const int d = EI[(long long)kE + e];
  const float4 v = *(const float4*)(X + (size_t)s * 64 + c4);
  float* p = AGG + (size_t)d * 64 + c4;
  atomicAdd(p + 0, v.x);
  atomicAdd(p + 1, v.y);
  atomicAdd(p + 2, v.z);
  atomicAdd(p + 3, v.w);
}

// =====================================================================
// Graph-wise LayerNorm (+ReLU, optional residual): one block per graph.
// Xout = (Res?Res:0) + relu( (Hin - mean)/sqrt(var+eps) * w + b )
// mean/var over all 512*64 values of the graph.
// =====================================================================
__global__ void graphln_kernel(const float* __restrict__ Hin,
                               const float* __restrict__ Res,  // may be null
                               float* __restrict__ Xout,
                               const float* __restrict__ wP,
                               const float* __restrict__ bP, int li) {
  const int g = blockIdx.x;
  const float* h = Hin + (size_t)g * kNPG * kH;
  float sum = 0.0f, sq = 0.0f;
  for (int i = threadIdx.x; i < kNPG * kH; i += 256) {
    const float v = h[i];
    sum += v;
    sq += v * v;
  }
  __shared__ float s1[256], s2[256];
  s1[threadIdx.x] = sum;
  s2[threadIdx.x] = sq;
  __syncthreads();
  for (int st = 128; st > 0; st >>= 1) {
    if (threadIdx.x < st) {
      s1[threadIdx.x] += s1[threadIdx.x + st];
      s2[threadIdx.x] += s2[threadIdx.x + st];
    }
    __syncthreads();
  }
  const float inv_n = 1.0f / (float)(kNPG * kH);
  const float mean = s1[0] * inv_n;
  const float var = s2[0] * inv_n - mean * mean;
  const float inv = rsqrtf(var + 1e-5f);
  const float w = wP[li], bb = bP[li];
  float* xo = Xout + (size_t)g * kNPG * kH;
  if (Res) {
    const float* rs = Res + (size_t)g * kNPG * kH;
    for (int i = threadIdx.x; i < kNPG * kH; i += 256) {
      float v = (h[i] - mean) * inv * w + bb;
      v = v > 0.0f ? v : 0.0f;
      xo[i] = rs[i] + v;
    }
  } else {
    for (int i = threadIdx.x; i < kNPG * kH; i += 256) {
      float v = (h[i] - mean) * inv * w + bb;
      xo[i] = v > 0.0f ? v : 0.0f;
    }
  }
}

// =====================================================================
// Per-graph channel sum: GB[b, l, :] = sum_n X[b, n, :]
// =====================================================================
__global__ void segsum_kernel(const float* __restrict__ X,
                              float* __restrict__ GB, int l) {
  const int g = blockIdx.x;
  const int ch = threadIdx.x & 63;
  const int part = threadIdx.x >> 6;  // 0..3
  const float* xb = X + (size_t)g * kNPG * kH;
  float s = 0.0f;
  for (int n = part * 128; n < part * 128 + 128; ++n)
    s += xb[(size_t)n * 64 + ch];
  __shared__ float sm[256];
  sm[threadIdx.x] = s;
  __syncthreads();
  if (threadIdx.x < 64) {
    const float t = sm[threadIdx.x] + sm[threadIdx.x + 64] +
                    sm[threadIdx.x + 128] + sm[threadIdx.x + 192];
    GB[((size_t)g * (kL + 1) + l) * 64 + ch] = t;
  }
}

// =====================================================================
// 1x1 conv (4->12) + leaky ReLU
// =====================================================================
__global__ void conv1_lrelu(const float* __restrict__ IN,
                            const float* __restrict__ W,
                            const float* __restrict__ Bi,
                            float* __restrict__ OUT) {
  const int b = blockIdx.x / 12, co = blockIdx.x % 12;
  const float* ib = IN + (size_t)b * 4 * 4096;
  const float w0 = W[co * 4 + 0], w1 = W[co * 4 + 1];
  const float w2 = W[co * 4 + 2], w3 = W[co * 4 + 3];
  const float bias = Bi[co];
  float* ob = OUT + ((size_t)b * 12 + co) * 4096;
  for (int p = threadIdx.x; p < 4096; p += 256) {
    float v = w0 * ib[p] + w1 * ib[4096 + p] + w2 * ib[2 * 4096 + p] +
              w3 * ib[3 * 4096 + p] + bias;
    ob[p] = v >= 0.0f ? v : 0.01f * v;
  }
}

// =====================================================================
// Instance norm over 64x64 per (b,c). grid = B*C
// =====================================================================
__global__ void inorm_kernel(const float* __restrict__ IN,
                             float* __restrict__ OUT,
                             const float* __restrict__ W,
                             const float* __restrict__ Bi, int C) {
  const int b = blockIdx.x / C, c = blockIdx.x % C;
  const float* ib = IN + ((size_t)b * C + c) * 4096;
  float* ob = OUT + ((size_t)b * C + c) * 4096;
  float sum = 0.0f, sq = 0.0f;
  for (int i = threadIdx.x; i < 4096; i += 256) {
    const float v = ib[i];
    sum += v;
    sq += v * v;
  }
  __shared__ float s1[256], s2[256];
  s1[threadIdx.x] = sum;
  s2[threadIdx.x] = sq;
  __syncthreads();
  for (int st = 128; st > 0; st >>= 1) {
    if (threadIdx.x < st) {
      s1[threadIdx.x] += s1[threadIdx.x + st];
      s2[threadIdx.x] += s2[threadIdx.x + st];
    }
    __syncthreads();
  }
  const float mean = s1[0] * (1.0f / 4096.0f);
  const float var = s2[0] * (1.0f / 4096.0f) - mean * mean;
  const float inv = rsqrtf(var + 1e-5f);
  const float w = W[c], bb = Bi[c];
  for (int i = threadIdx.x; i < 4096; i += 256)
    ob[i] = (ib[i] - mean) * inv * w + bb;
}

// =====================================================================
// 3x3 conv (12 -> Cout), pad 1, optional residual, (leaky) ReLU.
// grid = B*Cout, 256 threads.
// =====================================================================
__global__ void conv3x3_kernel(const float* __restrict__ IN,
                               const float* __restrict__ W,
                               const float* __restrict__ Bi,
                               float* __restrict__ OUT, int Cout,
                               int use_res, float negslope) {
  const int b = blockIdx.x / Cout, co = blockIdx.x % Cout;
  const float* ib = IN + (size_t)b * 12 * 4096;
  float* ob = OUT + ((size_t)b * Cout + co) * 4096;
  const float bias = Bi[co];
  for (int p = threadIdx.x; p < 4096; p += 256) {
    const int y = p >> 6, x = p & 63;
    float acc = bias;
    for (int ci = 0; ci < 12; ++ci) {
      const float* ip = ib + (size_t)ci * 4096;
      const float* wp = W + ((size_t)co * 12 + ci) * 9;
#pragma unroll
      for (int ky = 0; ky < 3; ++ky) {
        const int yy = y + ky - 1;
        if (yy < 0 || yy > 63) continue;
#pragma unroll
        for (int kx = 0; kx < 3; ++kx) {
          const int xx = x + kx - 1;
          if (xx < 0 || xx > 63) continue;
          acc += wp[ky * 3 + kx] * ip[yy * 64 + xx];
        }
      }
    }
    if (use_res) acc += ib[(size_t)co * 4096 + p];
    ob[p] = acc >= 0.0f ? acc : negslope * acc;
  }
}

// =====================================================================
// Final MLP: feat=[gbI|gbJ|nbf] (4608) -> 32 -> LN -> relu -> 1
// one block (256 threads) per graph.
// =====================================================================
__global__ void final_kernel(const float* __restrict__ gbI,
                             const float* __restrict__ gbJ,
                             const float* __restrict__ NBF,
                             const float* __restrict__ r1w,
                             const float* __restrict__ r1b,
                             const float* __restrict__ lnw,
                             const float* __restrict__ lnb,
                             const float* __restrict__ r2w,
                             const float* __restrict__ r2b,
                             float* __restrict__ out) {
  const int b = blockIdx.x;
  const int o = threadIdx.x >> 3;  // 0..31
  const int s = threadIdx.x & 7;
  const float* wrow = r1w + (size_t)o * 4608;
  float acc = 0.0f;
  for (int j = s; j < 4608; j += 8) {
    float f;
    if (j < 256)
      f = gbI[(size_t)b * 256 + j];
    else if (j < 512)
      f = gbJ[(size_t)b * 256 + (j - 256)];
    else
      f = NBF[(size_t)b * 4096 + (j - 512)];
    acc += f * wrow[j];
  }
  __shared__ float sm[256];
  __shared__ float s1[32];
  sm[threadIdx.x] = acc;
  __syncthreads();
  if (s == 0) {
    float t = 0.0f;
#pragma unroll
    for (int k = 0; k < 8; ++k) t += sm[o * 8 + k];
    s1[o] = t + r1b[o];
  }
  __syncthreads();
  if (threadIdx.x == 0) {
    float m = 0.0f;
    for (int k = 0; k < 32; ++k) m += s1[k];
    m *= (1.0f / 32.0f);
    float v = 0.0f;
    for (int k = 0; k < 32; ++k) {
      const float d = s1[k] - m;
      v += d * d;
    }
    v *= (1.0f / 32.0f);
    const float inv = rsqrtf(v + 1e-5f);
    float r = 0.0f;
    for (int k = 0; k < 32; ++k) {
      float t = (s1[k] - m) * inv * lnw[k] + lnb[k];
      t = t > 0.0f ? t : 0.0f;
      r += t * r2w[k];
    }
    out[b] = r + r2b[0];
  }
}

// =====================================================================
extern "C" void kernel_launch(void* const* d_in, const int* in_sizes, int n_in,
                              void* d_out, int out_size, void* d_ws,
                              size_t ws_size, hipStream_t stream) {
  (void)in_sizes; (void)n_in; (void)out_size; (void)ws_size;

  const float* x_t     = (const float*)d_in[0];
  const float* x_s     = (const float*)d_in[1];
  const int*   ei_t    = (const int*)d_in[2];
  const int*   ei_s    = (const int*)d_in[3];
  // d_in[4], d_in[5] = batch (implicit: node n -> graph n/512)
  const float* emb_w   = (const float*)d_in[6];
  const float* emb_b   = (const float*)d_in[7];
  const float* gin_w   = (const float*)d_in[8];
  const float* gin_b   = (const float*)d_in[9];
  const float* gin_eps = (const float*)d_in[10];
  const float* ln1_w   = (const float*)d_in[11];
  const float* ln1_b   = (const float*)d_in[12];
  const float* ffn_w   = (const float*)d_in[13];
  const float* ffn_b   = (const float*)d_in[14];
  const float* ln2_w   = (const float*)d_in[15];
  const float* ln2_b   = (const float*)d_in[16];
  const float* c1_w    = (const float*)d_in[17];
  const float* c1_b    = (const float*)d_in[18];
  const float* in1_w   = (const float*)d_in[19];
  const float* in1_b   = (const float*)d_in[20];
  const float* c2_w    = (const float*)d_in[21];
  const float* c2_b    = (const float*)d_in[22];
  const float* in2_w   = (const float*)d_in[23];
  const float* in2_b   = (const float*)d_in[24];
  const float* c3_w    = (const float*)d_in[25];
  const float* c3_b    = (const float*)d_in[26];
  const float* in3_w   = (const float*)d_in[27];
  const float* in3_b   = (const float*)d_in[28];
  const float* c4_w    = (const float*)d_in[29];
  const float* c4_b    = (const float*)d_in[30];
  const float* in4_w   = (const float*)d_in[31];
  const float* in4_b   = (const float*)d_in[32];
  const float* r1_w    = (const float*)d_in[33];
  const float* r1_b    = (const float*)d_in[34];
  const float* lnr_w   = (const float*)d_in[35];
  const float* lnr_b   = (const float*)d_in[36];
  const float* r2_w    = (const float*)d_in[37];
  const float* r2_b    = (const float*)d_in[38];

  // ---------------- workspace layout (floats) ----------------
  float* ws = (float*)d_ws;
  size_t off = 0;
  float* xb[2]; xb[0] = ws + off; off += kNM; xb[1] = ws + off; off += kNM;
  float* agg  = ws + off; off += kNM;
  float* hbuf = ws + off; off += kNM;
  const size_t nbSz = (size_t)kB * (kL + 1) * 4096;  // 2,097,152
  float* nbb[2]; nbb[0] = ws + off; off += nbSz; nbb[1] = ws + off; off += nbSz;
  float* nbc = ws + off; off += nbSz;
  const size_t tSz = (size_t)kB * 12 * 4096;
  float* t0 = ws + off; off += tSz;
  float* t1 = ws + off; off += tSz;
  float* nbf = ws + off; off += (size_t)kB * 4096;
  float* gbb[2];
  gbb[0] = ws + off; off += (size_t)kB * (kL + 1) * 64;
  gbb[1] = ws + off; off += (size_t)kB * (kL + 1) * 64;

  // ---------------- backbone (both sides) ----------------
  for (int side = 0; side < 2; ++side) {
    const float* xin = side ? x_s : x_t;
    const int* ei = side ? ei_s : ei_t;
    float* x = xb[side];

    // embedding: x = xin[N,32] @ emb_w^T + emb_b
    gemm_rows64_t<kIN, false><<<kN / 32, 256, 0, stream>>>(
        xin, nullptr, nullptr, 0, emb_w, emb_b, x);
    segsum_kernel<<<kB, 256, 0, stream>>>(x, gbb[side], 0);
    gram_kernel<<<kB, 512, 0, stream>>>(x, nbb[side], 0);

    for (int l = 0; l < kL; ++l) {
      hipMemsetAsync(agg, 0, kNM * sizeof(float), stream);
      edge_agg<<<(kE * 16) / 256, 256, 0, stream>>>(x, ei, agg);
      // h = ((1+eps)*x + agg) @ gin_w[l]^T + gin_b[l]
      gemm_rows64_t<kH, true><<<kN / 32, 256, 0, stream>>>(
          x, agg, gin_eps, l, gin_w + (size_t)l * kH * kH,
          gin_b + (size_t)l * kH, hbuf);
      // x = x + relu(graph_ln(h))
      graphln_kernel<<<kB, 256, 0, stream>>>(hbuf, x, x, ln1_w, ln1_b, l);
      // x2 = x @ ffn_w[l]^T + ffn_b[l]
      gemm_rows64_t<kH, false><<<kN / 32, 256, 0, stream>>>(
          x, nullptr, nullptr, 0, ffn_w + (size_t)l * kH * kH,
          ffn_b + (size_t)l * kH, hbuf);
      // x = relu(graph_ln(x2))
      graphln_kernel<<<kB, 256, 0, stream>>>(hbuf, nullptr, x, ln2_w, ln2_b, l);
      segsum_kernel<<<kB, 256, 0, stream>>>(x, gbb[side], l + 1);
      gram_kernel<<<kB, 512, 0, stream>>>(x, nbb[side], l + 1);
    }
  }

  // ---------------- Gram product + CNN head ----------------
  nbcomb_kernel<<<kB * (kL + 1), 512, 0, stream>>>(nbb[0], nbb[1], nbc);

  conv1_lrelu<<<kB * 12, 256, 0, stream>>>(nbc, c1_w, c1_b, t0);
  inorm_kernel<<<kB * 12, 256, 0, stream>>>(t0, t1, in1_w, in1_b, 12);

  conv3x3_kernel<<<kB * 12, 256, 0, stream>>>(t1, c2_w, c2_b, t0, 12, 1, 0.01f);
  inorm_kernel<<<kB * 12, 256, 0, stream>>>(t0, t1, in2_w, in2_b, 12);

  conv3x3_kernel<<<kB * 12, 256, 0, stream>>>(t1, c3_w, c3_b, t0, 12, 1, 0.01f);
  inorm_kernel<<<kB * 12, 256, 0, stream>>>(t0, t1, in3_w, in3_b, 12);

  conv3x3_kernel<<<kB * 1, 256, 0, stream>>>(t1, c4_w, c4_b, t0, 1, 0, 0.0f);
  inorm_kernel<<<kB * 1, 256, 0, stream>>>(t0, nbf, in4_w, in4_b, 1);

  // ---------------- final MLP ----------------
  final_kernel<<<kB, 256, 0, stream>>>(gbb[0], gbb[1], nbf, r1_w, r1_b, lnr_w,
                                       lnr_b, r2_w, r2_b, (float*)d_out);
}